// Model_pro_14817637171533
// MI455X (gfx1250) — compile-verified
//
#include <hip/hip_runtime.h>

typedef __attribute__((ext_vector_type(2))) float v2f;
typedef __attribute__((ext_vector_type(8))) float v8f;

#define NATOMS 2048
#define NG     48
#define NPTS   (48 * 48 * 48)          /* 110592 */
#define GRIDSP 0.5f
#define E2F    7.3890560989306495f     /* e^2 */
#define NINE_OVER_E2 1.21801754912951f /* 9 / e^2 */
#define LOG2E  1.44269504088896340736f

// ---------------------------------------------------------------------------
// Prep: fold VDW table into per-atom constants (SoA in workspace).
//   P[i] = { x, y, z, -2*log2e/r^2 }            (exp core: f1 = exp2(k1*d2))
//   Q[i] = { 4/(e2 r2), -12/(e2 r), r2, (float)ch }   (shell, threshold, chan)
// ---------------------------------------------------------------------------
__global__ void prep_kernel(const float* __restrict__ coords,
                            const int* __restrict__ chidx,
                            float4* __restrict__ P,
                            float4* __restrict__ Q) {
  int i = blockIdx.x * blockDim.x + threadIdx.x;
  if (i >= NATOMS) return;
  const float vdw[8] = {1.7f, 1.55f, 1.52f, 1.8f, 1.95f, 1.8f, 1.4f, 1.7f};
  float x = coords[3 * i + 0];
  float y = coords[3 * i + 1];
  float z = coords[3 * i + 2];
  int c = chidx[i] & 7;
  float r = vdw[c];
  float r2 = r * r;
  P[i] = make_float4(x, y, z, -2.0f * LOG2E / r2);
  Q[i] = make_float4(4.0f / (E2F * r2), -12.0f / (E2F * r), r2, (float)c);
}

// ---------------------------------------------------------------------------
// Per-pair density, fully branchless. Raw v_sqrt_f32 / v_exp_f32 (TRANS32,
// co-executes with VALU). The piecewise select is a mask BLEND so both sides
// are unconditionally consumed -> clang cannot if-convert to divergent
// control flow (the round-2 regression).
// ---------------------------------------------------------------------------
__device__ __forceinline__ float eval_density(float ax, float ay, float az,
                                              float k1, float qa, float qb,
                                              float r2,
                                              float px, float py, float pz) {
  float dx = ax - px, dy = ay - py, dz = az - pz;
  float d2 = fmaf(dx, dx, fmaf(dy, dy, dz * dz));
  float d  = __builtin_amdgcn_sqrtf(d2);
  float f1 = __builtin_amdgcn_exp2f(d2 * k1);                 // exp(-2 d2/r2)
  float f2 = fmaf(qa, d2, fmaf(qb, d, NINE_OVER_E2));         // quadratic shell
  float inCore  = d2 < r2 ? 1.0f : 0.0f;
  float inShell = d2 < 2.25f * r2 ? 1.0f : 0.0f;              // includes core
  // core: f1 ; shell: f2 ; outside: 0
  return fmaf(inCore, f1, (inShell - inCore) * f2);
}

__device__ __forceinline__ float4 sel4(bool c, float4 a, float4 b) {
  return make_float4(c ? a.x : b.x, c ? a.y : b.y, c ? a.z : b.z, c ? a.w : b.w);
}

// One 4-atom WMMA step: A = one-hot[16ch x 4atoms], B = f[4atoms x 16pts].
// 32-bit operand layout (ISA 7.12.2): v0 lanes0-15 = K0, lanes16-31 = K2;
//                                     v1 lanes0-15 = K1, lanes16-31 = K3.
// So lanes<16 evaluate atoms base+0/base+1; lanes>=16 evaluate base+2/base+3.
__device__ __forceinline__ v8f wmma_step(int base, bool up, float rowf,
                                         float px, float py, float pz,
                                         const float4* __restrict__ P,
                                         const float4* __restrict__ Q,
                                         v8f acc) {
  float4 p0 = P[base + 0], p1 = P[base + 1], p2 = P[base + 2], p3 = P[base + 3];
  float4 q0 = Q[base + 0], q1 = Q[base + 1], q2 = Q[base + 2], q3 = Q[base + 3];

  float4 pA = sel4(up, p2, p0), pB = sel4(up, p3, p1);
  float4 qA = sel4(up, q2, q0), qB = sel4(up, q3, q1);

  float fA = eval_density(pA.x, pA.y, pA.z, pA.w, qA.x, qA.y, qA.z, px, py, pz);
  float fB = eval_density(pB.x, pB.y, pB.z, pB.w, qB.x, qB.y, qB.z, px, py, pz);

  v2f Bv = {fA, fB};
  v2f Av = {rowf == qA.w ? 1.0f : 0.0f, rowf == qB.w ? 1.0f : 0.0f};

  // D = onehot x f + C : segment-sum into channel rows via the matrix pipe.
  return __builtin_amdgcn_wmma_f32_16x16x4_f32(false, Av, false, Bv,
                                               (short)0, acc, false, false);
}

// ---------------------------------------------------------------------------
// One wave32 per tile of 16 consecutive z grid points. 6912 tiles total.
// D-matrix layout: lanes 0-15 hold column p, rows (channels) 0..7 in v[0..7].
// ---------------------------------------------------------------------------
__global__ __launch_bounds__(256) void splat_kernel(const float4* __restrict__ P,
                                                    const float4* __restrict__ Q,
                                                    float* __restrict__ out) {
  const unsigned tid  = threadIdx.x;
  const unsigned lane = tid & 31u;
  const unsigned wave = tid >> 5;
  const unsigned tile = blockIdx.x * 8u + wave;   // 0..6911 = 48*48*3
  const unsigned zt = tile % 3u;                  // z-tile within line
  const unsigned xy = tile / 3u;
  const unsigned gy = xy % 48u;
  const unsigned gx = xy / 48u;
  const unsigned col = lane & 15u;                // grid-point column (N)
  const bool up = lane >= 16u;
  const float rowf = (float)col;                  // one-hot A row (M), exact

  const float px = GRIDSP * (float)gx;
  const float py = GRIDSP * (float)gy;
  const float pz = GRIDSP * (float)(zt * 16u + col);

  v8f acc0 = {};
  v8f acc1 = {};
  for (int base = 0; base < NATOMS; base += 8) {
    acc0 = wmma_step(base + 0, up, rowf, px, py, pz, P, Q, acc0);
    acc1 = wmma_step(base + 4, up, rowf, px, py, pz, P, Q, acc1);
  }
  v8f acc = acc0 + acc1;

  if (lane < 16u) {
    const unsigned pidx = (gx * 48u + gy) * 48u + zt * 16u + col;
#pragma unroll
    for (int ch = 0; ch < 8; ++ch)
      out[(unsigned)ch * (unsigned)NPTS + pidx] = acc[ch];
  }
}

// ---------------------------------------------------------------------------
extern "C" void kernel_launch(void* const* d_in, const int* in_sizes, int n_in,
                              void* d_out, int out_size, void* d_ws, size_t ws_size,
                              hipStream_t stream) {
  const float* coords = (const float*)d_in[0];  // [2048,3] f32
  const int*   chidx  = (const int*)d_in[1];    // [2048] i32
  float* out = (float*)d_out;                   // [8,48,48,48] f32

  float4* P = (float4*)d_ws;                              // 32 KB
  float4* Q = (float4*)((char*)d_ws + NATOMS * 16);       // 32 KB

  prep_kernel<<<(NATOMS + 255) / 256, 256, 0, stream>>>(coords, chidx, P, Q);
  splat_kernel<<<(NG * NG * 3) / 8, 256, 0, stream>>>(P, Q, out);
}